// LinePooling_89120571392672
// MI455X (gfx1250) — compile-verified
//
#include <hip/hip_runtime.h>

#define ALIGN_SIZE 128
#define SPATIAL_SCALE 0.25f
#define CH 256
#define IMG_H 128
#define IMG_W 128
#define HW (IMG_H * IMG_W)

__device__ __forceinline__ void make_corner(int xi, int yi, float w,
                                            int& off, float& wgt) {
    // Reference: gather with clipped index, then zero the value if the
    // (unclipped) corner is out of range. Values are finite, so folding the
    // validity into the weight is exactly equivalent.
    const bool v = (xi >= 0) && (xi < IMG_W) && (yi >= 0) && (yi < IMG_H);
    int xc = xi < 0 ? 0 : (xi > IMG_W - 1 ? IMG_W - 1 : xi);
    int yc = yi < 0 ? 0 : (yi > IMG_H - 1 ? IMG_H - 1 : yi);
    off = yc * IMG_W + xc;
    wgt = v ? w : 0.0f;
}

__global__ __launch_bounds__(256) void line_pool_kernel(
    const float* __restrict__ feat,
    const float* __restrict__ coord_st,
    const float* __restrict__ coord_ed,
    float* __restrict__ out,
    int num_ed) {
    const int j    = blockIdx.x;          // ed index
    const int i    = blockIdx.y;          // st index
    const int tid  = threadIdx.x;
    const int lane = tid & 31;            // wave32 lane
    const int co   = tid >> 5;            // channel phase 0..7 (8 waves)

    const int bs = (int)coord_st[0];
    const float* __restrict__ img = feat + (size_t)bs * CH * HW;

    const float stx = coord_st[i * 3 + 1] * SPATIAL_SCALE;
    const float sty = coord_st[i * 3 + 2] * SPATIAL_SCALE;
    const float edx = coord_ed[j * 3 + 1] * SPATIAL_SCALE;
    const float edy = coord_ed[j * 3 + 2] * SPATIAL_SCALE;

    // Per-lane sample metadata: lane L covers k = L, L+32, L+64, L+96.
    int   off[4][4];
    float wgt[4][4];
    const float step = 1.0f / (float)(ALIGN_SIZE - 1);
#pragma unroll
    for (int m = 0; m < 4; ++m) {
        const int   k = lane + 32 * m;
        const float t = (float)k * step;
        const float x = stx + (edx - stx) * t;
        const float y = sty + (edy - sty) * t;
        // Replicate reference normalize -> denormalize round trip (f32).
        const float gx = x / (float)(IMG_W - 1) * 2.0f - 1.0f;
        const float gy = y / (float)(IMG_H - 1) * 2.0f - 1.0f;
        const float xs = (gx + 1.0f) * 0.5f * (float)(IMG_W - 1);
        const float ys = (gy + 1.0f) * 0.5f * (float)(IMG_H - 1);
        const float x0f = floorf(xs);
        const float y0f = floorf(ys);
        const float wx1 = xs - x0f, wx0 = 1.0f - wx1;
        const float wy1 = ys - y0f, wy0 = 1.0f - wy1;
        const int x0 = (int)x0f, y0 = (int)y0f;
        const int x1 = x0 + 1,   y1 = y0 + 1;
        make_corner(x0, y0, wx0 * wy0, off[m][0], wgt[m][0]);
        make_corner(x1, y0, wx1 * wy0, off[m][1], wgt[m][1]);
        make_corner(x0, y1, wx0 * wy1, off[m][2], wgt[m][2]);
        make_corner(x1, y1, wx1 * wy1, off[m][3], wgt[m][3]);
    }

    float* __restrict__ outb =
        out + (((size_t)i * num_ed + j) * CH) * ALIGN_SIZE;

    // Channel loop: 16 L2 gathers + 4 interp + 4 coalesced NT stores / iter.
#pragma unroll 2
    for (int c = co; c < CH; c += 8) {
        const float* __restrict__ plane = img + (size_t)c * HW;

        // gfx1250 global_prefetch_b8: warm the next channel plane's two
        // gather rows per sample while this iteration's loads are in flight.
        if (c + 8 < CH) {
            const float* pn = plane + 8 * HW;
#pragma unroll
            for (int m = 0; m < 4; ++m) {
                __builtin_prefetch(pn + off[m][0], 0, 3);  // y0 row
                __builtin_prefetch(pn + off[m][2], 0, 3);  // y1 row
            }
        }

        float acc[4];
#pragma unroll
        for (int m = 0; m < 4; ++m) {
            const float v0 = plane[off[m][0]];
            const float v1 = plane[off[m][1]];
            const float v2 = plane[off[m][2]];
            const float v3 = plane[off[m][3]];
            // Same term order as reference: (x0,y0),(x1,y0),(x0,y1),(x1,y1)
            acc[m] = v0 * wgt[m][0] + v1 * wgt[m][1] +
                     v2 * wgt[m][2] + v3 * wgt[m][3];
        }

        // Output is 512 MB of write-once data: store with the non-temporal
        // hint (th:TH_STORE_NT) so it rinses through L2 without evicting the
        // L2-resident 16 MB feature image that feeds the gathers.
        float* __restrict__ o = outb + (size_t)c * ALIGN_SIZE + lane;
#pragma unroll
        for (int m = 0; m < 4; ++m)
            __builtin_nontemporal_store(acc[m], o + 32 * m);  // 128B/wave
    }
}

extern "C" void kernel_launch(void* const* d_in, const int* in_sizes, int n_in,
                              void* d_out, int out_size, void* d_ws,
                              size_t ws_size, hipStream_t stream) {
    const float* feat = (const float*)d_in[0];
    const float* cst  = (const float*)d_in[1];
    const float* ced  = (const float*)d_in[2];
    float* out        = (float*)d_out;

    const int num_st = in_sizes[1] / 3;
    const int num_ed = in_sizes[2] / 3;

    dim3 grid(num_ed, num_st);  // one block per (st, ed) line
    line_pool_kernel<<<grid, 256, 0, stream>>>(feat, cst, ced, out, num_ed);
}